// layer_simple_26113401160397
// MI455X (gfx1250) — compile-verified
//
#include <hip/hip_runtime.h>
#include <stddef.h>

typedef __attribute__((ext_vector_type(2))) float v2f;
typedef __attribute__((ext_vector_type(8))) float v8f;

#define BS   8
#define NF   64
#define NN   2048
#define JJ   3
#define NOUT 64
#define NCH  (JJ * NF)    // 192 input channels to conv
#define NCH2 (2 * NOUT)   // 128 output channels (concat of cv2, cv1 paths)
#define EPSV 1e-5f

// workspace layout (in floats)
#define OFF_X1T 0                              // x1 transposed: [b][m][ch]  (8*2048*192)
#define SZ_X1T  (BS * NN * NCH)
#define OFF_ZB  (OFF_X1T + SZ_X1T)            // zb pre-norm:  [b][o][m]    (8*128*2048)
#define SZ_ZB   (BS * NCH2 * NN)
#define OFF_CWC (OFF_ZB + SZ_ZB)              // combined conv weights [o][ch] (128*192)
#define SZ_CWC  (NCH2 * NCH)
#define OFF_CB  (OFF_CWC + SZ_CWC)            // combined bias (128)
#define OFF_SUM (OFF_CB + NCH2)               // per-channel sum   (128)
#define OFF_SQ  (OFF_SUM + NCH2)              // per-channel sumsq (128)
#define OFF_SS  (OFF_SQ + NCH2)               // scale(128) then shift(128)

// ---------------------------------------------------------------------------
// K0: build combined conv weights/bias (zb = concat[cv2-path, cv1-path]) and
//     zero the BN accumulators. 24576 threads exactly covers NCH2*NCH.
// ---------------------------------------------------------------------------
__global__ void prep_kernel(const float* __restrict__ cv1w, const float* __restrict__ cv1b,
                            const float* __restrict__ cv2w, const float* __restrict__ cv2b,
                            float* __restrict__ ws) {
    int tid = blockIdx.x * blockDim.x + threadIdx.x;
    if (tid < NCH2 * NCH) {
        int o = tid / NCH, ch = tid % NCH;
        ws[OFF_CWC + tid] = (o < NOUT) ? cv2w[o * NCH + ch] : cv1w[(o - NOUT) * NCH + ch];
    }
    if (tid < NCH2) ws[OFF_CB + tid] = (tid < NOUT) ? cv2b[tid] : cv1b[tid - NOUT];
    if (tid < 2 * NCH2) ws[OFF_SUM + tid] = 0.0f;   // sum + sumsq are contiguous
}

// ---------------------------------------------------------------------------
// K1: x1t[b][m][ch] = sum_n W[b,m,n,j] * X[b,f,n]   (ch = j*64+f)
// Wave32 WMMA f32 16x16x4. Each wave owns 16 m-rows, all 64 f, all 3 j
// (12 v8f accumulators). One lane streams one W row (contiguous 24B per
// 4-n step covering all j) -> W read exactly once, and we pass it straight
// through to d_out (tuple output includes W), saving a separate 402MB copy.
// ---------------------------------------------------------------------------
__global__ __launch_bounds__(256) void gemm_graph_kernel(const float* __restrict__ W,
                                                         const float* __restrict__ X,
                                                         float* __restrict__ ws,
                                                         float* __restrict__ w_out) {
    const int wave = threadIdx.x >> 5;
    const int lane = threadIdx.x & 31;
    const int row  = lane & 15;     // A-matrix M index within tile (lanes 0-15 / 16-31)
    const int half = lane >> 4;     // selects K pair {0,1} vs {2,3}
    const int b    = blockIdx.x / (NN / 128);
    const int mt   = blockIdx.x % (NN / 128);
    const int m    = mt * 128 + wave * 16 + row;

    const float* __restrict__ Wrow = W + ((size_t)b * NN + m) * (size_t)(NN * JJ);
    float*       __restrict__ Orow = w_out + ((size_t)b * NN + m) * (size_t)(NN * JJ);
    const float* __restrict__ Xb   = X + (size_t)b * NF * NN;

    v8f acc[JJ][4] = {};

    for (int k = 0; k < NN; k += 4) {
        const int n0 = k + 2 * half;                  // this half-wave's K pair start
        const float* wp = Wrow + (size_t)n0 * JJ;     // 6 contiguous floats: (n0,j0..2),(n0+1,j0..2)
        v2f w01 = *(const v2f*)(wp);
        v2f w23 = *(const v2f*)(wp + 2);
        v2f w45 = *(const v2f*)(wp + 4);
        // pass W through to the output tuple while it is in registers
        float* op = Orow + (size_t)n0 * JJ;
        *(v2f*)(op)     = w01;
        *(v2f*)(op + 2) = w23;
        *(v2f*)(op + 4) = w45;

        const float a6[6] = { w01.x, w01.y, w23.x, w23.y, w45.x, w45.y };

        v2f xv[4];
#pragma unroll
        for (int t = 0; t < 4; ++t)
            xv[t] = *(const v2f*)(Xb + (size_t)(t * 16 + row) * NN + n0);  // B: K=n0..n0+1, f=t*16+row

#pragma unroll
        for (int j = 0; j < JJ; ++j) {
            v2f a;
            a.x = a6[j];        // A vgpr0: K even slot
            a.y = a6[3 + j];    // A vgpr1: K odd slot
#pragma unroll
            for (int t = 0; t < 4; ++t)
                acc[j][t] = __builtin_amdgcn_wmma_f32_16x16x4_f32(
                    false, a, false, xv[t], (short)0, acc[j][t], false, false);
        }
    }

    // D layout: vgpr r, lanes0-15 -> (M=r, N=lane); lanes16-31 -> (M=r+8, N=lane-16)
    float* __restrict__ x1t = ws + OFF_X1T;
#pragma unroll
    for (int j = 0; j < JJ; ++j)
#pragma unroll
        for (int t = 0; t < 4; ++t)
#pragma unroll
            for (int r = 0; r < 8; ++r) {
                const int mm = mt * 128 + wave * 16 + r + 8 * half;
                const int ch = j * NF + t * 16 + row;
                x1t[((size_t)b * NN + mm) * NCH + ch] = acc[j][t][r];
            }
}

// ---------------------------------------------------------------------------
// K2: zb[b][o][m] = relu(cwc[o,:] . x1t[b][m][:] + cb[o]), plus masked
// per-channel partial sums for BN (LDS float atomics -> global atomics).
// WMMA: M=node tile(16), N=out-channel tile(16), K=192.
// ---------------------------------------------------------------------------
__global__ __launch_bounds__(256) void conv_bn_stats_kernel(const float* __restrict__ mask,
                                                            float* __restrict__ ws) {
    __shared__ float s_sum[NCH2];
    __shared__ float s_sq[NCH2];

    const int wave = threadIdx.x >> 5;
    const int lane = threadIdx.x & 31;
    const int row  = lane & 15;
    const int half = lane >> 4;
    const int b    = blockIdx.x / (NN / 128);
    const int mt   = blockIdx.x % (NN / 128);
    const int m    = mt * 128 + wave * 16 + row;

    if (threadIdx.x < NCH2) { s_sum[threadIdx.x] = 0.0f; s_sq[threadIdx.x] = 0.0f; }
    __syncthreads();

    const float* __restrict__ x1row = ws + OFF_X1T + ((size_t)b * NN + m) * NCH;
    const float* __restrict__ cwc   = ws + OFF_CWC;

    v8f acc[8] = {};
    for (int k = 0; k < NCH; k += 4) {
        const int k0 = k + 2 * half;
        v2f a = *(const v2f*)(x1row + k0);                       // A: M=node, K=ch (contiguous)
#pragma unroll
        for (int t = 0; t < 8; ++t) {
            v2f bf = *(const v2f*)(cwc + (size_t)(t * 16 + row) * NCH + k0);  // B: K=ch, N=o
            acc[t] = __builtin_amdgcn_wmma_f32_16x16x4_f32(
                false, a, false, bf, (short)0, acc[t], false, false);
        }
    }

    const float* __restrict__ cb  = ws + OFF_CB;
    float*       __restrict__ zb  = ws + OFF_ZB;
    const float* __restrict__ mkb = mask + (size_t)b * NN;

#pragma unroll
    for (int t = 0; t < 8; ++t) {
        const int o = t * 16 + row;
        const float bias = cb[o];
        float ps = 0.0f, pq = 0.0f;
#pragma unroll
        for (int r = 0; r < 8; ++r) {
            const int mm = mt * 128 + wave * 16 + r + 8 * half;
            float v = acc[t][r] + bias;
            v = v > 0.0f ? v : 0.0f;
            const float mk = mkb[mm];
            zb[((size_t)b * NCH2 + o) * NN + mm] = v;
            ps += v * mk;
            pq += v * v * mk;
        }
        atomicAdd(&s_sum[o], ps);
        atomicAdd(&s_sq[o], pq);
    }
    __syncthreads();
    if (threadIdx.x < NCH2) {
        atomicAdd(ws + OFF_SUM + threadIdx.x, s_sum[threadIdx.x]);
        atomicAdd(ws + OFF_SQ + threadIdx.x, s_sq[threadIdx.x]);
    }
}

// ---------------------------------------------------------------------------
// K3: per-channel mean/var -> (scale, shift)
// ---------------------------------------------------------------------------
__global__ void bn_finalize_kernel(const float* __restrict__ Nb, const float* __restrict__ gamma,
                                   const float* __restrict__ beta, float* __restrict__ ws) {
    const int o = threadIdx.x;
    if (o >= NCH2) return;
    float total = 0.0f;
    for (int i = 0; i < BS; ++i) total += Nb[i];
    const float mean = ws[OFF_SUM + o] / total;
    const float var  = ws[OFF_SQ + o] / total - mean * mean;
    const float sc   = gamma[o] * rsqrtf(var + EPSV);
    ws[OFF_SS + o]        = sc;
    ws[OFF_SS + NCH2 + o] = beta[o] - mean * sc;
}

// ---------------------------------------------------------------------------
// K4: zbn = (zb * scale + shift) * mask -> d_out (reference layout b,128,N)
// ---------------------------------------------------------------------------
__global__ void bn_apply_kernel(const float* __restrict__ mask, const float* __restrict__ ws,
                                float* __restrict__ out) {
    const int idx = blockIdx.x * blockDim.x + threadIdx.x;
    if (idx >= BS * NCH2 * NN) return;
    const int n = idx % NN;
    const int o = (idx / NN) % NCH2;
    const int b = idx / (NN * NCH2);
    const float v = ws[OFF_ZB + idx];
    out[idx] = (v * ws[OFF_SS + o] + ws[OFF_SS + NCH2 + o]) * mask[(size_t)b * NN + n];
}

// ---------------------------------------------------------------------------
extern "C" void kernel_launch(void* const* d_in, const int* in_sizes, int n_in,
                              void* d_out, int out_size, void* d_ws, size_t ws_size,
                              hipStream_t stream) {
    const float* X     = (const float*)d_in[0];
    const float* W     = (const float*)d_in[1];
    const float* Nb    = (const float*)d_in[2];
    const float* mask  = (const float*)d_in[3];
    const float* cv1w  = (const float*)d_in[4];
    const float* cv1b  = (const float*)d_in[5];
    const float* cv2w  = (const float*)d_in[6];
    const float* cv2b  = (const float*)d_in[7];
    const float* gamma = (const float*)d_in[8];
    const float* beta  = (const float*)d_in[9];

    float* out = (float*)d_out;
    float* ws  = (float*)d_ws;
    float* w_out = out + (size_t)BS * NCH2 * NN;   // tuple slot 2: W pass-through

    prep_kernel<<<(NCH2 * NCH + 255) / 256, 256, 0, stream>>>(cv1w, cv1b, cv2w, cv2b, ws);
    gemm_graph_kernel<<<BS * (NN / 128), 256, 0, stream>>>(W, X, ws, w_out);
    conv_bn_stats_kernel<<<BS * (NN / 128), 256, 0, stream>>>(mask, ws);
    bn_finalize_kernel<<<1, NCH2, 0, stream>>>(Nb, gamma, beta, ws);
    bn_apply_kernel<<<(BS * NCH2 * NN + 255) / 256, 256, 0, stream>>>(mask, ws, out);
}